// Encoder_54107997995610
// MI455X (gfx1250) — compile-verified
//
#include <hip/hip_runtime.h>

typedef __attribute__((ext_vector_type(2))) float v2f;
typedef __attribute__((ext_vector_type(8))) float v8f;

#define NF 128          // feature width (NFEAT == NHID == 128)
#define WSTRIDE 132     // padded LDS stride (bank stride 4 -> conflict-free)

// ---------------------------------------------------------------------------
// support = X @ W  using V_WMMA_F32_16X16X4_F32.
// Block: 256 threads = 8 waves. Block owns 64 rows; wave w owns cols [16w,16w+16).
// W (128x128 f32, 64KB) staged transposed into LDS once per block.
// ---------------------------------------------------------------------------
__global__ __launch_bounds__(256) void gcn_gemm_wmma(
    const float* __restrict__ X, const float* __restrict__ W,
    float* __restrict__ OUT, int nrows)
{
    __shared__ float Wt[NF * WSTRIDE];   // Wt[n*WSTRIDE + k] = W[k*128 + n]

    const int tid  = threadIdx.x;
    const int lane = tid & 31;
    const int wave = tid >> 5;
    const int row0 = blockIdx.x * 64;

    // Cooperative transpose-load of W: coalesced global reads, scattered LDS writes.
    for (int i = tid; i < NF * NF; i += 256) {
        const int k = i >> 7;      // row of W (K)
        const int n = i & 127;     // col of W (N)
        Wt[n * WSTRIDE + k] = W[i];
    }
    __syncthreads();

    const int l16   = lane & 15;
    const int lhalf = lane >> 4;            // 0: K+0/K+1, 1: K+2/K+3
    const int n0    = wave * 16;
    const float* brow = &Wt[(n0 + l16) * WSTRIDE + lhalf * 2];

    for (int mt = 0; mt < 4; ++mt) {
        const int arow  = row0 + mt * 16 + l16;
        const int arowc = arow < nrows ? arow : (nrows - 1);   // clamp: no OOB read
        const float* aptr = X + (size_t)arowc * NF + lhalf * 2;

        v8f c = {};
#pragma unroll
        for (int k0 = 0; k0 < NF; k0 += 4) {
            // A frag: lanes 0-15 -> {K0,K1}, lanes 16-31 -> {K2,K3}  (8B-aligned b64)
            v2f a = *(const v2f*)(aptr + k0);
            // B frag: same K split, N = n0 + l16, read from transposed LDS (b64)
            v2f b = *(const v2f*)(brow + k0);
            c = __builtin_amdgcn_wmma_f32_16x16x4_f32(
                    /*neg_a=*/false, a, /*neg_b=*/false, b,
                    /*c_mod=*/(short)0, c, /*reuse_a=*/false, /*reuse_b=*/false);
        }

        // C/D layout: VGPR r, lanes 0-15 -> M=r, lanes 16-31 -> M=8+r; N = lane%16
        const int mbase = row0 + mt * 16 + lhalf * 8;
#pragma unroll
        for (int r = 0; r < 8; ++r) {
            const int m = mbase + r;
            if (m < nrows) OUT[(size_t)m * NF + n0 + l16] = c[r];
        }
    }
}

// ---------------------------------------------------------------------------
// agg[dst[e]] += w[e] * support[src[e]]  — one wave per edge (grid-stride).
// 32 lanes x float4 = one full 512B feature row per gather; f32 atomic scatter.
// Both buffers are L2-resident (25.6 MB each vs 192 MB L2).
// ---------------------------------------------------------------------------
__global__ __launch_bounds__(256) void gcn_scatter(
    const float* __restrict__ S, const int* __restrict__ src,
    const int* __restrict__ dst, const float* __restrict__ ew,
    float* __restrict__ agg, int nedges)
{
    const int lane   = threadIdx.x & 31;
    const int gwave  = (int)((blockIdx.x * blockDim.x + threadIdx.x) >> 5);
    const int nwaves = (int)((gridDim.x * blockDim.x) >> 5);

    for (int e = gwave; e < nedges; e += nwaves) {
        const int   s = src[e];
        const int   d = dst[e];
        const float w = ew[e];
        const float4 v = *(const float4*)(S + (size_t)s * NF + lane * 4);
        float* o = agg + (size_t)d * NF + lane * 4;
        atomicAdd(o + 0, v.x * w);
        atomicAdd(o + 1, v.y * w);
        atomicAdd(o + 2, v.z * w);
        atomicAdd(o + 3, v.w * w);
    }
}

// ---------------------------------------------------------------------------
// out = relu(in + bias), vectorized float4 (128 cols = 32 float4 per row).
// ---------------------------------------------------------------------------
__global__ __launch_bounds__(256) void gcn_bias_relu(
    const float* __restrict__ in, const float* __restrict__ bias,
    float* __restrict__ out, int n4)
{
    const int i = blockIdx.x * blockDim.x + threadIdx.x;
    if (i >= n4) return;
    float4 v = ((const float4*)in)[i];
    const float4 b = ((const float4*)bias)[i & 31];
    v.x = fmaxf(v.x + b.x, 0.0f);
    v.y = fmaxf(v.y + b.y, 0.0f);
    v.z = fmaxf(v.z + b.z, 0.0f);
    v.w = fmaxf(v.w + b.w, 0.0f);
    ((float4*)out)[i] = v;
}

// ---------------------------------------------------------------------------
extern "C" void kernel_launch(void* const* d_in, const int* in_sizes, int n_in,
                              void* d_out, int out_size, void* d_ws, size_t ws_size,
                              hipStream_t stream)
{
    const float* x  = (const float*)d_in[0];   // [N,128]
    const int*   ei = (const int*)  d_in[1];   // [2,E]
    const float* ew = (const float*)d_in[2];   // [E]
    const float* W1 = (const float*)d_in[3];   // [128,128]
    const float* b1 = (const float*)d_in[4];   // [128]
    const float* W2 = (const float*)d_in[5];   // [128,128]
    const float* b2 = (const float*)d_in[6];   // [128]
    float* out = (float*)d_out;

    const int N = in_sizes[0] / NF;            // 50000
    const int E = in_sizes[2];                 // 800000
    const int* src = ei;
    const int* dst = ei + E;

    float* buf0 = (float*)d_ws;                // N*128 f32 (25.6 MB)
    float* buf1 = buf0 + (size_t)N * NF;       // N*128 f32 (25.6 MB)
    const size_t featBytes = (size_t)N * NF * sizeof(float);

    const dim3 gGemm((N + 63) / 64);
    const int  n4     = N * NF / 4;
    const dim3 gElem((n4 + 255) / 256);
    const dim3 gScat(2048);
    const dim3 blk(256);

    // ---- Layer 1 ----
    gcn_gemm_wmma<<<gGemm, blk, 0, stream>>>(x, W1, buf0, N);           // support1
    hipMemsetAsync(buf1, 0, featBytes, stream);
    gcn_scatter<<<gScat, blk, 0, stream>>>(buf0, src, dst, ew, buf1, E); // agg1
    gcn_bias_relu<<<gElem, blk, 0, stream>>>(buf1, b1, buf0, n4);        // h1

    // ---- Layer 2 ----
    gcn_gemm_wmma<<<gGemm, blk, 0, stream>>>(buf0, W2, buf1, N);         // support2
    hipMemsetAsync(buf0, 0, featBytes, stream);
    gcn_scatter<<<gScat, blk, 0, stream>>>(buf1, src, dst, ew, buf0, E); // agg2
    gcn_bias_relu<<<gElem, blk, 0, stream>>>(buf0, b2, out, n4);         // output
}